// Net_GAT_36335423324386
// MI455X (gfx1250) — compile-verified
//
#include <hip/hip_runtime.h>
#include <math.h>

#define NN       100000      // nodes
#define NE       1600000     // raw edges
#define ETOT     (NE + NN)   // with self loops
#define NG       512         // graphs
#define NEG_SLP  0.2f

typedef __attribute__((ext_vector_type(2))) float v2f;
typedef __attribute__((ext_vector_type(8))) float v8f;

// ---------------------------------------------------------------- utilities
__device__ __forceinline__ float atomicMaxF32(float* addr, float value) {
    if (value >= 0.0f)
        return __int_as_float(atomicMax((int*)addr, __float_as_int(value)));
    else
        return __uint_as_float(atomicMin((unsigned int*)addr, __float_as_uint(value)));
}

__global__ void fill_kernel(float* __restrict__ p, float v, int n) {
    int i = blockIdx.x * blockDim.x + threadIdx.x;
    if (i < n) p[i] = v;
}

// ------------------------------------------------- f32 WMMA GEMM (16-row tiles)
// C[n_rows, NC] = A[n_rows, K] @ B[K, NC]  (+bias) (relu)
// One wave per 16-row tile; sweeps NC/16 column tiles; K stepped by 4 via
// V_WMMA_F32_16X16X4_F32.  n_rows must be a multiple of 16.
__global__ void gemm_wmma_f32(const float* __restrict__ A, const float* __restrict__ B,
                              const float* __restrict__ bias, float* __restrict__ C,
                              int n_rows, int K, int NC, int relu) {
    int wave = (blockIdx.x * blockDim.x + threadIdx.x) >> 5;
    int lane = threadIdx.x & 31;
    int row0 = wave * 16;
    if (row0 >= n_rows) return;            // whole-wave uniform exit: EXEC stays all-1s

    int mrow  = lane & 15;                 // M index (A) / N index (B, C)
    int khalf = lane >> 4;                 // 0: K pair {0,1}, 1: K pair {2,3}
    int ntiles = NC >> 4;

    for (int nt = 0; nt < ntiles; ++nt) {
        int ncol = nt * 16 + mrow;
        v8f acc = {};
        for (int kb = 0; kb < K; kb += 4) {
            int k0 = kb + khalf * 2;
            v2f a, b;
            a.x = A[(size_t)(row0 + mrow) * K + k0];
            a.y = A[(size_t)(row0 + mrow) * K + k0 + 1];
            b.x = B[(size_t)k0 * NC + ncol];
            b.y = B[(size_t)(k0 + 1) * NC + ncol];
            acc = __builtin_amdgcn_wmma_f32_16x16x4_f32(
                false, a, false, b, (short)0, acc, false, false);
        }
        // C/D layout: VGPR r -> M = r (lanes 0-15) / r+8 (lanes 16-31), N = lane&15
        for (int r = 0; r < 8; ++r) {
            int row = row0 + r + khalf * 8;
            float v = acc[r];
            if (bias) v += bias[ncol];
            if (relu) v = v > 0.0f ? v : 0.0f;
            C[(size_t)row * NC + ncol] = v;
        }
    }
}

// ------------------------------------------------- attention scalar projections
// as[n,h] = sum_c h[n,h,c]*a_src[h,c] ; ad likewise
__global__ void alpha_kernel(const float* __restrict__ h,
                             const float* __restrict__ a_src, const float* __restrict__ a_dst,
                             float* __restrict__ as, float* __restrict__ ad,
                             int n, int H, int C) {
    int i = blockIdx.x * blockDim.x + threadIdx.x;
    if (i >= n * H) return;
    int node = i / H, hh = i - node * H;
    const float* hp = h + (size_t)node * H * C + hh * C;
    float ss = 0.0f, sd = 0.0f;
    for (int c = 0; c < C; ++c) {
        float v = hp[c];
        ss += v * a_src[hh * C + c];
        sd += v * a_dst[hh * C + c];
    }
    as[i] = ss;
    ad[i] = sd;
}

// ------------------------------------------------- edge passes
__device__ __forceinline__ void edge_sd(int i, const int* __restrict__ ei, int& s, int& d) {
    if (i < NE) { s = ei[i]; d = ei[NE + i]; }
    else        { s = d = i - NE; }           // self loops appended
}

__global__ void attn_max_kernel(const int* __restrict__ ei,
                                const float* __restrict__ as, const float* __restrict__ ad,
                                float* __restrict__ m, int H) {
    int i = blockIdx.x * blockDim.x + threadIdx.x;
    if (i >= ETOT) return;
    int s, d; edge_sd(i, ei, s, d);
    for (int h = 0; h < H; ++h) {
        float e = as[s * H + h] + ad[d * H + h];
        e = e > 0.0f ? e : NEG_SLP * e;
        atomicMaxF32(&m[d * H + h], e);
    }
}

__global__ void attn_sum_kernel(const int* __restrict__ ei,
                                const float* __restrict__ as, const float* __restrict__ ad,
                                const float* __restrict__ m, float* __restrict__ dn, int H) {
    int i = blockIdx.x * blockDim.x + threadIdx.x;
    if (i >= ETOT) return;
    int s, d; edge_sd(i, ei, s, d);
    for (int h = 0; h < H; ++h) {
        float e = as[s * H + h] + ad[d * H + h];
        e = e > 0.0f ? e : NEG_SLP * e;
        atomicAdd(&dn[d * H + h], expf(e - m[d * H + h]));
    }
}

__global__ void attn_agg_kernel(const int* __restrict__ ei,
                                const float* __restrict__ as, const float* __restrict__ ad,
                                const float* __restrict__ m, const float* __restrict__ dn,
                                const float* __restrict__ hin, float* __restrict__ out,
                                int H, int C) {
    int i = blockIdx.x * blockDim.x + threadIdx.x;
    if (i >= ETOT) return;
    int s, d; edge_sd(i, ei, s, d);
    for (int h = 0; h < H; ++h) {
        float e = as[s * H + h] + ad[d * H + h];
        e = e > 0.0f ? e : NEG_SLP * e;
        float alpha = expf(e - m[d * H + h]) / (dn[d * H + h] + 1e-16f);
        const float* sp = hin + (size_t)s * H * C + h * C;
        float*       op = out + (size_t)d * H * C + h * C;
        for (int c = 0; c < C; ++c)
            atomicAdd(&op[c], sp[c] * alpha);
    }
}

// ------------------------------------------------- pointwise / pooling / head
__global__ void bias_relu_kernel(float* __restrict__ p, const float* __restrict__ b,
                                 int n, int F) {
    int i = blockIdx.x * blockDim.x + threadIdx.x;
    if (i >= n * F) return;
    float v = p[i] + b[i % F];
    p[i] = v > 0.0f ? v : 0.0f;
}

__global__ void pool_kernel(const float* __restrict__ h, const int* __restrict__ batch,
                            float* __restrict__ pooled, int n, int F) {
    int i = blockIdx.x * blockDim.x + threadIdx.x;
    if (i >= n * F) return;
    int node = i / F, f = i - node * F;
    atomicAdd(&pooled[(size_t)batch[node] * F + f], h[i]);
}

__global__ void head_kernel(const float* __restrict__ hidden, const float* __restrict__ lw2,
                            const float* __restrict__ lb2, float* __restrict__ out) {
    int i = blockIdx.x * blockDim.x + threadIdx.x;
    if (i >= NG * 2) return;
    int g = i >> 1, o = i & 1;
    float s = lb2[o];
    for (int k = 0; k < 64; ++k)
        s += hidden[g * 64 + k] * lw2[k * 2 + o];
    out[i] = 1.0f / (1.0f + expf(-s));
}

// ---------------------------------------------------------------- launcher
static inline int cdiv(long n, int b) { return (int)((n + b - 1) / b); }

extern "C" void kernel_launch(void* const* d_in, const int* in_sizes, int n_in,
                              void* d_out, int out_size, void* d_ws, size_t ws_size,
                              hipStream_t stream) {
    const float* x    = (const float*)d_in[0];
    const int*   ei   = (const int*)  d_in[1];
    const int*   batch= (const int*)  d_in[2];
    const float* W1   = (const float*)d_in[3];
    const float* aS1  = (const float*)d_in[4];
    const float* aD1  = (const float*)d_in[5];
    const float* b1   = (const float*)d_in[6];
    const float* W2   = (const float*)d_in[7];
    const float* aS2  = (const float*)d_in[8];
    const float* aD2  = (const float*)d_in[9];
    const float* b2   = (const float*)d_in[10];
    const float* lw1  = (const float*)d_in[11];
    const float* lb1  = (const float*)d_in[12];
    const float* lw2  = (const float*)d_in[13];
    const float* lb2  = (const float*)d_in[14];
    float* out = (float*)d_out;

    // workspace carve-up (floats)
    float* ws   = (float*)d_ws;
    float* h1   = ws;                    // NN*32
    float* out1 = h1   + (size_t)NN*32;  // NN*32 (aggregated, then relu'd in place)
    float* h2   = out1 + (size_t)NN*32;  // NN*32
    float* out2 = h2   + (size_t)NN*32;  // NN*32
    float* as1  = out2 + (size_t)NN*32;  // NN*2
    float* ad1  = as1  + (size_t)NN*2;
    float* m1   = ad1  + (size_t)NN*2;
    float* dn1  = m1   + (size_t)NN*2;
    float* as2  = dn1  + (size_t)NN*2;   // NN
    float* ad2  = as2  + NN;
    float* m2   = ad2  + NN;
    float* dn2  = m2   + NN;
    float* pooled = dn2 + NN;            // NG*32
    float* hidden = pooled + NG*32;      // NG*64

    const int B = 256;
    const dim3 blk(B);

    // ---- layer 1 linear: h1 = x @ W1   [NN,64]x[64,32]
    {
        int thr = (NN / 16) * 32;
        gemm_wmma_f32<<<cdiv(thr, B), blk, 0, stream>>>(x, W1, nullptr, h1, NN, 64, 32, 0);
    }
    alpha_kernel<<<cdiv((long)NN*2, B), blk, 0, stream>>>(h1, aS1, aD1, as1, ad1, NN, 2, 16);

    fill_kernel<<<cdiv((long)NN*2, B), blk, 0, stream>>>(m1, -INFINITY, NN*2);
    fill_kernel<<<cdiv((long)NN*2, B), blk, 0, stream>>>(dn1, 0.0f, NN*2);
    fill_kernel<<<cdiv((long)NN*32, B), blk, 0, stream>>>(out1, 0.0f, NN*32);

    attn_max_kernel<<<cdiv(ETOT, B), blk, 0, stream>>>(ei, as1, ad1, m1, 2);
    attn_sum_kernel<<<cdiv(ETOT, B), blk, 0, stream>>>(ei, as1, ad1, m1, dn1, 2);
    attn_agg_kernel<<<cdiv(ETOT, B), blk, 0, stream>>>(ei, as1, ad1, m1, dn1, h1, out1, 2, 16);
    bias_relu_kernel<<<cdiv((long)NN*32, B), blk, 0, stream>>>(out1, b1, NN, 32);

    // ---- layer 2 linear: h2 = out1 @ W2   [NN,32]x[32,32]
    {
        int thr = (NN / 16) * 32;
        gemm_wmma_f32<<<cdiv(thr, B), blk, 0, stream>>>(out1, W2, nullptr, h2, NN, 32, 32, 0);
    }
    alpha_kernel<<<cdiv((long)NN, B), blk, 0, stream>>>(h2, aS2, aD2, as2, ad2, NN, 1, 32);

    fill_kernel<<<cdiv((long)NN, B), blk, 0, stream>>>(m2, -INFINITY, NN);
    fill_kernel<<<cdiv((long)NN, B), blk, 0, stream>>>(dn2, 0.0f, NN);
    fill_kernel<<<cdiv((long)NN*32, B), blk, 0, stream>>>(out2, 0.0f, NN*32);

    attn_max_kernel<<<cdiv(ETOT, B), blk, 0, stream>>>(ei, as2, ad2, m2, 1);
    attn_sum_kernel<<<cdiv(ETOT, B), blk, 0, stream>>>(ei, as2, ad2, m2, dn2, 1);
    attn_agg_kernel<<<cdiv(ETOT, B), blk, 0, stream>>>(ei, as2, ad2, m2, dn2, h2, out2, 1, 32);
    bias_relu_kernel<<<cdiv((long)NN*32, B), blk, 0, stream>>>(out2, b2, NN, 32);

    // ---- pooling
    fill_kernel<<<cdiv((long)NG*32, B), blk, 0, stream>>>(pooled, 0.0f, NG*32);
    pool_kernel<<<cdiv((long)NN*32, B), blk, 0, stream>>>(out2, batch, pooled, NN, 32);

    // ---- MLP head: hidden = relu(pooled @ lw1 + lb1)   [NG,32]x[32,64]
    {
        int thr = (NG / 16) * 32;
        gemm_wmma_f32<<<cdiv(thr, B), blk, 0, stream>>>(pooled, lw1, lb1, hidden, NG, 32, 64, 1);
    }
    head_kernel<<<cdiv((long)NG*2, B), blk, 0, stream>>>(hidden, lw2, lb2, out);
}